// CDistLoss_12962211299684
// MI455X (gfx1250) — compile-verified
//
#include <hip/hip_runtime.h>
#include <hip/hip_bf16.h>

#define NPTS   4096
#define DFEAT  128
#define MPOS   (NPTS - 1)      // positions after dropping self
#define SORT_T 512

typedef float v2f __attribute__((ext_vector_type(2)));
typedef float v8f __attribute__((ext_vector_type(8)));

// ---------------------------------------------------------------------------
// Phase 0: squared row norms  sq[i] = sum_k x[i][k]^2
// ---------------------------------------------------------------------------
__global__ void sqnorm_kernel(const float* __restrict__ x, float* __restrict__ sq) {
  int i = blockIdx.x * blockDim.x + threadIdx.x;
  if (i >= NPTS) return;
  const float4* p = (const float4*)(x + (size_t)i * DFEAT);
  float s = 0.f;
#pragma unroll
  for (int k = 0; k < DFEAT / 4; ++k) {
    float4 v = p[k];
    s = fmaf(v.x, v.x, s); s = fmaf(v.y, v.y, s);
    s = fmaf(v.z, v.z, s); s = fmaf(v.w, v.w, s);
  }
  sq[i] = s;
}

// ---------------------------------------------------------------------------
// Phase 1: D[i][j] = sqrt(max(sq[i]+sq[j]-2*x_i.x_j, 1e-12))
// One wave (32 threads) per 16x16 tile, fp32 WMMA 16x16x4, K-loop over 128.
//
// A fragment (16x4 f32): lanes 0-15 -> M=lane,    K = k0+{0,1}
//                        lanes16-31 -> M=lane-16, K = k0+{2,3}
// B fragment (4x16 f32): lanes 0-15 -> N=lane,    K = k0+{0,1}
//                        lanes16-31 -> N=lane-16, K = k0+{2,3}
// Both are contiguous K-pairs per lane -> single b64 load each.
// C/D (16x16 f32, 8 VGPRs): VGPR r -> M=r (lanes 0-15), M=r+8 (lanes 16-31)
// ---------------------------------------------------------------------------
__global__ void gram_dist_kernel(const float* __restrict__ x,
                                 const float* __restrict__ sq,
                                 float* __restrict__ D) {
  const int jt   = blockIdx.x;
  const int it   = blockIdx.y;
  const int lane = threadIdx.x;        // 0..31, full wave, EXEC all ones
  const int half = lane >> 4;          // 0 or 1
  const int l16  = lane & 15;
  const int i0   = it * 16;
  const int j0   = jt * 16;

  const float* arow = x + (size_t)(i0 + l16) * DFEAT + half * 2;
  const float* brow = x + (size_t)(j0 + l16) * DFEAT + half * 2;

  // hint the full K rows into cache (global_prefetch_b8)
  __builtin_prefetch(arow, 0, 1);
  __builtin_prefetch(arow + 64, 0, 1);
  __builtin_prefetch(brow, 0, 1);
  __builtin_prefetch(brow + 64, 0, 1);

  v8f c = {};
#pragma unroll
  for (int k0 = 0; k0 < DFEAT; k0 += 4) {
    v2f a = *(const v2f*)(arow + k0);
    v2f b = *(const v2f*)(brow + k0);
    c = __builtin_amdgcn_wmma_f32_16x16x4_f32(
        /*neg_a=*/false, a, /*neg_b=*/false, b,
        /*c_mod=*/(short)0, c, /*reuse_a=*/false, /*reuse_b=*/false);
  }

  const float sqj = sq[j0 + l16];
#pragma unroll
  for (int r = 0; r < 8; ++r) {
    const int m  = r + half * 8;
    float d2 = sq[i0 + m] + sqj - 2.0f * c[r];
    D[(size_t)(i0 + m) * NPTS + j0 + l16] = sqrtf(fmaxf(d2, 1e-12f));
  }
}

// ---------------------------------------------------------------------------
// Phase 2: one workgroup per row.
//   key[j] = (float_bits(D[row][j]) << 32) | j   (stable ascending argsort)
//   bitonic sort in LDS, drop position 0 (self), blocked scan of agree bits,
//   hardness-weighted score + performance sums.
// ---------------------------------------------------------------------------
__global__ void __launch_bounds__(SORT_T)
row_score_kernel(const float* __restrict__ D, const int* __restrict__ y,
                 float* __restrict__ score, float* __restrict__ perf) {
  __shared__ unsigned long long s_key[NPTS];   // 32 KB
  __shared__ float              s_cum[NPTS];   // 16 KB
  __shared__ int                s_ts[SORT_T];  //  2 KB
  __shared__ float              s_ra[SORT_T];  //  2 KB
  __shared__ float              s_rb[SORT_T];  //  2 KB

  const int row = blockIdx.x;
  const int tid = threadIdx.x;
  const int myy = y[row];

  // build sort keys
  for (int j = tid; j < NPTS; j += SORT_T) {
    float dd = D[(size_t)row * NPTS + j];
    unsigned int db = __float_as_uint(dd);   // dd >= 0 -> monotone as uint
    s_key[j] = ((unsigned long long)db << 32) | (unsigned int)j;
  }
  __syncthreads();

  // bitonic sort, ascending
  for (int k = 2; k <= NPTS; k <<= 1) {
    for (int j = k >> 1; j > 0; j >>= 1) {
      for (int i = tid; i < NPTS; i += SORT_T) {
        int l = i ^ j;
        if (l > i) {
          unsigned long long a = s_key[i];
          unsigned long long b = s_key[l];
          bool swap = ((i & k) == 0) ? (a > b) : (a < b);
          if (swap) { s_key[i] = b; s_key[l] = a; }
        }
      }
      __syncthreads();
    }
  }

  // blocked inclusive scan of agree bits (position 0 = self, forced to 0)
  const int base = tid * (NPTS / SORT_T);   // 8 consecutive positions/thread
  int vals[NPTS / SORT_T];
  int c0 = 0;
#pragma unroll
  for (int u = 0; u < NPTS / SORT_T; ++u) {
    int p  = base + u;
    int jj = (int)(s_key[p] & 0xFFFFFFFFull);
    int a  = (p == 0) ? 0 : ((y[jj] == myy) ? 1 : 0);
    c0 += a;
    vals[u] = c0;
  }
  s_ts[tid] = c0;
  __syncthreads();
  for (int off = 1; off < SORT_T; off <<= 1) {
    int add = (tid >= off) ? s_ts[tid - off] : 0;
    __syncthreads();
    s_ts[tid] += add;
    __syncthreads();
  }
  const int excl = s_ts[tid] - c0;
#pragma unroll
  for (int u = 0; u < NPTS / SORT_T; ++u)
    s_cum[base + u] = (float)(vals[u] + excl);
  __syncthreads();

  const float Sa = s_cum[NPTS - 1];
  const float Sd = (float)MPOS - Sa;          // cum_d[p] = p - cum_a[p]
  const float invSa = 1.0f / (Sa + 1e-7f);
  const float invSd = 1.0f / (Sd + 1e-7f);

  float accS = 0.f, accP = 0.f;
  for (int p = tid; p < NPTS; p += SORT_T) {
    if (p == 0) continue;                     // drop self
    unsigned long long kk = s_key[p];
    float dist = __uint_as_float((unsigned int)(kk >> 32));
    int   jj   = (int)(kk & 0xFFFFFFFFull);
    float a    = (y[jj] == myy) ? 1.0f : 0.0f;
    float d    = 1.0f - a;
    float ca   = s_cum[p];
    float fa   = ca * invSa;                          // found_agree
    float fd   = ((float)p - ca) * invSd;             // found_disagree
    float ih   = 1.0f / (fa + 0.1f);                  // HARDNESS_RATIO
    float negh = d * fd * ih;
    float posh = a * (fd + 0.1f) * ih;
    accS += fmaxf(dist * posh - dist * negh, 0.0f);   // MARGIN = 0
    accP += fa - fd;
  }

  __syncthreads();
  s_ra[tid] = accS;
  s_rb[tid] = accP;
  __syncthreads();
  for (int s = SORT_T / 2; s > 0; s >>= 1) {
    if (tid < s) { s_ra[tid] += s_ra[tid + s]; s_rb[tid] += s_rb[tid + s]; }
    __syncthreads();
  }
  if (tid == 0) {
    score[row] = s_ra[0] / (float)MPOS;   // mean over N-1 positions
    perf[row]  = s_rb[0];
  }
}

// ---------------------------------------------------------------------------
// Phase 3: global min of perf, then out[i] = score[i] / (perf[i]-min+1)
// ---------------------------------------------------------------------------
__global__ void __launch_bounds__(1024)
finalize_kernel(const float* __restrict__ score, const float* __restrict__ perf,
                float* __restrict__ out) {
  __shared__ float mn[1024];
  const int tid = threadIdx.x;
  float m = 3.0e38f;
  for (int i = tid; i < NPTS; i += 1024) m = fminf(m, perf[i]);
  mn[tid] = m;
  __syncthreads();
  for (int s = 512; s > 0; s >>= 1) {
    if (tid < s) mn[tid] = fminf(mn[tid], mn[tid + s]);
    __syncthreads();
  }
  const float minperf = mn[0];
  for (int i = tid; i < NPTS; i += 1024)
    out[i] = score[i] / ((perf[i] - minperf) + 1.0f);
}

// ---------------------------------------------------------------------------
extern "C" void kernel_launch(void* const* d_in, const int* in_sizes, int n_in,
                              void* d_out, int out_size, void* d_ws, size_t ws_size,
                              hipStream_t stream) {
  (void)in_sizes; (void)n_in; (void)out_size; (void)ws_size;
  const float* x = (const float*)d_in[0];
  const int*   y = (const int*)d_in[1];
  float* out = (float*)d_out;

  char*  ws  = (char*)d_ws;
  float* Dm  = (float*)ws;                                   // 64 MB (fits L2)
  float* sq  = (float*)(ws + (size_t)NPTS * NPTS * sizeof(float));
  float* sc  = sq + NPTS;
  float* pf  = sc + NPTS;

  sqnorm_kernel<<<NPTS / 256, 256, 0, stream>>>(x, sq);

  dim3 grid(NPTS / 16, NPTS / 16);
  gram_dist_kernel<<<grid, 32, 0, stream>>>(x, sq, Dm);

  row_score_kernel<<<NPTS, SORT_T, 0, stream>>>(Dm, y, sc, pf);

  finalize_kernel<<<1, 1024, 0, stream>>>(sc, pf, out);
}